// Stage2LagrangianStateSpaceModel_62234076119574
// MI455X (gfx1250) — compile-verified
//
#include <hip/hip_runtime.h>
#include <math.h>

// Kalman filter scan: T steps, state dim KD=256.
// Single 1024-thread workgroup (32 wave32s) carries the sequential scan;
// the three 256^3 f32 GEMMs per step run on V_WMMA_F32_16X16X4_F32.
// Each wave owns one 16-row tile stripe (8 column tiles) so one A fragment
// feeds 8 back-to-back WMMAs per k-substep. B panel is stored pair-interleaved
// in LDS so each B fragment is a single aligned ds_load_b64.

static constexpr int KD    = 256;   // state dimension
static constexpr int TPB   = 1024;  // threads per block = 32 waves
static constexpr int KSTEP = 16;    // staged K-panel width
static constexpr int LDA_S = 20;    // padded A-panel row stride (floats):
                                    // 16B-aligned float4 stores, conflict-free reads

typedef float v2f __attribute__((ext_vector_type(2)));
typedef float v8f __attribute__((ext_vector_type(8)));

__device__ __forceinline__ float sanitize(float x) {
    x = (x != x) ? 0.0f : x;                       // nan -> 0
    return fminf(fmaxf(x, -1.0e6f), 1.0e6f);       // +/-inf -> +/-CLIP, clip
}

// C = opA(A) * opB(B); A,B,C are KD x KD row-major f32 in global/L2.
// 16-wide K-panels staged through LDS with float4 vector loads.
// ldsB pair-interleaved: pair(grp,kbb,n) = {panel[4g+kbb][n], panel[4g+kbb+2][n]}
// at float offset grp*1024 + kbb*512 + 2n.
template<bool TA, bool TB>
__device__ void gemm_wmma(const float* __restrict__ A,
                          const float* __restrict__ B,
                          float* __restrict__ C,
                          float* ldsA, float* ldsB)
{
    const int tid  = threadIdx.x;
    const int lane = tid & 31;
    const int wave = tid >> 5;

    const int m0    = (wave >> 1) << 4;   // tile row (fixed per wave)
    const int nbase = (wave & 1) << 7;    // column half: 0 or 128

    // A-fragment layout (16x4 f32): lanes 0-15 -> K=0,1 ; lanes 16-31 -> K=2,3
    const int ma   = lane & 15;
    const int koff = (lane >> 4) << 1;
    // B-fragment layout (4x16 f32): VGPR0 rows K={0,1}, VGPR1 rows K={2,3}
    const int kb   = lane >> 4;
    const int nn   = lane & 15;

    // staging coordinates (4 floats per thread per panel per round)
    const int sar = tid >> 2;             // A panel row 0..255
    const int sak = (tid & 3) << 2;       // A panel k 0,4,8,12
    const int sbk = tid >> 6;             // B panel k 0..15
    const int sbc = (tid & 63) << 2;      // B panel col 0,4,...,252

    v8f acc[8] = {};

    #pragma unroll 1
    for (int k = 0; k < KD; k += KSTEP) {
        // ---- stage A panel: ldsA[r*LDA_S + kk] = opA(A)[r][k+kk] ----
        if (!TA) {
            const float4 v = *(const float4*)&A[sar * KD + k + sak];
            *(float4*)&ldsA[sar * LDA_S + sak] = v;
        } else {
            #pragma unroll
            for (int j = 0; j < 4; ++j)
                ldsA[sar * LDA_S + sak + j] = A[(k + sak + j) * KD + sar];
        }
        // ---- stage B panel (pair-interleaved) ----
        if (!TB) {
            // panel row sbk = 4*grp + j ; element -> pair(grp, j&1, c) slot (j>>1)
            const int grp  = sbk >> 2, j = sbk & 3;
            const float4 v = *(const float4*)&B[(k + sbk) * KD + sbc];
            float* dst = &ldsB[(grp << 10) + ((j & 1) << 9) + (sbc << 1) + (j >> 1)];
            dst[0] = v.x; dst[2] = v.y; dst[4] = v.z; dst[6] = v.w;
        } else {
            // panel[kk][c] = B[c][k+kk]; one column c, rows k4..k4+3 (one group)
            const int c  = tid >> 2;          // col 0..255
            const int k4 = (tid & 3) << 2;    // kk 0,4,8,12 -> grp = k4>>2
            const float4 v = *(const float4*)&B[c * KD + k + k4];
            float* base = &ldsB[((k4 >> 2) << 10) + (c << 1)];
            v2f p0; p0.x = v.x; p0.y = v.z;   // rows j=0 (slot0), j=2 (slot1)
            v2f p1; p1.x = v.y; p1.y = v.w;   // rows j=1 (slot0), j=3 (slot1)
            *(v2f*)&base[0]   = p0;
            *(v2f*)&base[512] = p1;
        }
        __syncthreads();

        #pragma unroll
        for (int g = 0; g < 4; ++g) {         // kk = 4*g
            v2f a;
            a.x = ldsA[(m0 + ma) * LDA_S + (g << 2) + koff];
            a.y = ldsA[(m0 + ma) * LDA_S + (g << 2) + koff + 1];
            v2f b[8];
            #pragma unroll
            for (int i = 0; i < 8; ++i) {
                const int n = nbase + (i << 4) + nn;
                b[i] = *(const v2f*)&ldsB[(g << 10) + (kb << 9) + (n << 1)];
            }
            #pragma unroll
            for (int i = 0; i < 8; ++i)
                acc[i] = __builtin_amdgcn_wmma_f32_16x16x4_f32(
                    false, a, false, b[i], (short)0, acc[i], false, false);
        }
        __syncthreads();
    }

    // C/D layout: VGPR r -> (M=r, N=lane) lanes 0-15 ; (M=r+8, N=lane-16)
    const int mb = m0 + ((lane >> 4) << 3);
    #pragma unroll
    for (int i = 0; i < 8; ++i) {
        const int n = nbase + (i << 4) + nn;
        #pragma unroll
        for (int r = 0; r < 8; ++r)
            C[(mb + r) * KD + n] = acc[i][r];
    }
}

__global__ __launch_bounds__(TPB)
void kf_scan_kernel(const float* __restrict__ obs,     // T*KD
                    const float* __restrict__ trans,   // T*KD*KD
                    const float* __restrict__ forc,    // T*KD
                    const float* __restrict__ pcov,    // KD*KD
                    const float* __restrict__ mcov,    // KD*KD
                    const float* __restrict__ imean,   // KD
                    const float* __restrict__ icovd,   // KD
                    float* __restrict__ out,           // 1 + 2*T*KD
                    float* __restrict__ ws,
                    int T)
{
    __shared__ __align__(16) float ldsA[KD * LDA_S];   // 20 KB
    __shared__ __align__(16) float ldsB[KSTEP * KD];   // 16 KB
    __shared__ float red[TPB];                         //  4 KB
    __shared__ float sh_diag;

    const int tid = threadIdx.x;

    // workspace (all KD x KD row-major f32 unless noted)
    float* P    = ws;                  // covariance carry
    float* Aw   = P    + KD * KD;      // sanitized A_t
    float* W1   = Aw   + KD * KD;      // scratch GEMM out
    float* W2   = W1   + KD * KD;      // scratch GEMM out
    float* PRED = W2   + KD * KD;      // pred_P (symmetric)
    float* Lm   = PRED + KD * KD;      // S -> Cholesky factor (lower)
    float* X    = Lm   + KD * KD;      // S^{-1} pred_P  (K = X^T)
    float* Qw   = X    + KD * KD;
    float* Rw   = Qw   + KD * KD;
    float* mv   = Rw   + KD * KD;      // mean carry        (KD)
    float* pm   = mv   + KD;           // predicted mean    (KD)
    float* inn  = pm   + KD;           // innovation        (KD)
    float* al   = inn  + KD;           // alpha             (KD)

    float* filt = out + 1;
    float* pred = out + 1 + (size_t)T * KD;

    // ---- one-time setup: Q, R (sanitize + symmetrize), P0, m0 ----
    #pragma unroll 1
    for (int e = tid; e < KD * KD; e += TPB) {
        const int i = e >> 8, j = e & 255;
        Qw[e] = 0.5f * (sanitize(pcov[i * KD + j]) + sanitize(pcov[j * KD + i]))
              + ((i == j) ? 1.0e-4f : 0.0f);
        Rw[e] = 0.5f * (sanitize(mcov[i * KD + j]) + sanitize(mcov[j * KD + i]));
        P[e]  = (i == j) ? fminf(fmaxf(icovd[i], 1.0e-6f), 1.0e6f) : 0.0f;
    }
    if (tid < KD) mv[tid] = sanitize(imean[tid]);
    float nll_acc = 0.0f;   // thread 0 only
    __syncthreads();

    #pragma unroll 1
    for (int t = 0; t < T; ++t) {
        const float* At = trans + (size_t)t * KD * KD;
        const float* yt = obs   + (size_t)t * KD;
        const float* ft = forc  + (size_t)t * KD;

        // sanitize A_t (L2-resident stream, 256 KB/step)
        #pragma unroll 1
        for (int e = tid; e < KD * KD; e += TPB) Aw[e] = sanitize(At[e]);
        // prefetch next step's transition block: one probe per 64B line
        if (t + 1 < T) {
            #pragma unroll 1
            for (int e = tid * 16; e < KD * KD; e += TPB * 16)
                __builtin_prefetch(At + KD * KD + e, 0, 1);
        }
        __syncthreads();

        // pred_m = Aw @ m + sanitize(f) ; innov = y - pred_m   (4 thr/row)
        {
            const int i = tid >> 2, q = tid & 3;
            float s = 0.0f;
            #pragma unroll 1
            for (int j = q * 64; j < q * 64 + 64; ++j) s += Aw[i * KD + j] * mv[j];
            red[tid] = s;
            __syncthreads();
            if (q == 0) {
                float p = red[tid] + red[tid + 1] + red[tid + 2] + red[tid + 3]
                        + sanitize(ft[i]);
                pm[i] = p;
                pred[(size_t)t * KD + i] = p;
                inn[i] = yt[i] - p;
            }
            __syncthreads();
        }

        // W1 = Aw @ P ; W2 = W1 @ Aw^T    (WMMA)
        gemm_wmma<false, false>(Aw, P, W1, ldsA, ldsB);
        __syncthreads();
        gemm_wmma<false, true>(W1, Aw, W2, ldsA, ldsB);
        __syncthreads();

        // PRED = symm(W2) + Q ; S = PRED + R + jitter*I  (into Lm)
        #pragma unroll 1
        for (int e = tid; e < KD * KD; e += TPB) {
            const int i = e >> 8, j = e & 255;
            const float p = 0.5f * (W2[e] + W2[j * KD + i]) + Qw[e];
            PRED[e] = p;
            Lm[e]   = p + Rw[e] + ((i == j) ? 1.0e-5f : 0.0f);
        }
        __syncthreads();

        // in-place Cholesky of Lm (lower); row-parallel column updates
        #pragma unroll 1
        for (int j = 0; j < KD; ++j) {
            float part = 0.0f;
            #pragma unroll 1
            for (int k = tid; k < j; k += TPB) part += Lm[j * KD + k] * Lm[j * KD + k];
            red[tid] = part;
            __syncthreads();
            #pragma unroll 1
            for (int s = TPB / 2; s > 0; s >>= 1) {
                if (tid < s) red[tid] += red[tid + s];
                __syncthreads();
            }
            if (tid == 0) {
                const float dv = sqrtf(fmaxf(Lm[j * KD + j] - red[0], 1.0e-12f));
                Lm[j * KD + j] = dv;
                sh_diag = dv;
            }
            __syncthreads();
            const float dj = sh_diag;
            #pragma unroll 1
            for (int i = j + 1 + tid; i < KD; i += TPB) {
                float s = Lm[i * KD + j];
                #pragma unroll 1
                for (int k = 0; k < j; ++k) s -= Lm[i * KD + k] * Lm[j * KD + k];
                Lm[i * KD + j] = s / dj;
            }
            __syncthreads();
        }

        // logdet (parallel) + alpha solve + nll (thread 0)
        {
            float part = 0.0f;
            #pragma unroll 1
            for (int i = tid; i < KD; i += TPB) part += logf(Lm[i * KD + i]);
            red[tid] = part;
            __syncthreads();
            #pragma unroll 1
            for (int s = TPB / 2; s > 0; s >>= 1) {
                if (tid < s) red[tid] += red[tid + s];
                __syncthreads();
            }
            if (tid == 0) {
                #pragma unroll 1
                for (int i = 0; i < KD; ++i) {            // L y = innov
                    float s = inn[i];
                    for (int k = 0; k < i; ++k) s -= Lm[i * KD + k] * al[k];
                    al[i] = s / Lm[i * KD + i];
                }
                #pragma unroll 1
                for (int i = KD - 1; i >= 0; --i) {       // L^T a = y
                    float s = al[i];
                    for (int k = i + 1; k < KD; ++k) s -= Lm[k * KD + i] * al[k];
                    al[i] = s / Lm[i * KD + i];
                }
                float quad = 0.0f;
                for (int i = 0; i < KD; ++i) quad += inn[i] * al[i];
                const float nll = 0.5f * (quad + 2.0f * red[0]
                                          + (float)KD * 1.8378770664093453f);
                nll_acc += fminf(nll, 1.0e4f);
            }
            __syncthreads();
        }

        // X = S^{-1} PRED : column-parallel fwd/bwd triangular solves
        if (tid < KD) {
            const int c = tid;
            #pragma unroll 1
            for (int i = 0; i < KD; ++i) {
                float s = PRED[i * KD + c];
                for (int k = 0; k < i; ++k) s -= Lm[i * KD + k] * X[k * KD + c];
                X[i * KD + c] = s / Lm[i * KD + i];
            }
            #pragma unroll 1
            for (int i = KD - 1; i >= 0; --i) {
                float s = X[i * KD + c];
                for (int k = i + 1; k < KD; ++k) s -= Lm[k * KD + i] * X[k * KD + c];
                X[i * KD + c] = s / Lm[i * KD + i];
            }
        }
        __syncthreads();

        // new_m = pm + X^T @ innov   (K @ innov with K = X^T)
        {
            const int i = tid >> 2, q = tid & 3;
            float s = 0.0f;
            #pragma unroll 1
            for (int j = q * 64; j < q * 64 + 64; ++j) s += X[j * KD + i] * inn[j];
            red[tid] = s;
            __syncthreads();
            if (q == 0) {
                const float nm = pm[i] + red[tid] + red[tid + 1] + red[tid + 2] + red[tid + 3];
                mv[i] = nm;
                filt[(size_t)t * KD + i] = nm;
            }
            __syncthreads();
        }

        // new_P = symm((I - K) PRED) = PRED - symm(X^T @ PRED)   (WMMA)
        gemm_wmma<true, false>(X, PRED, W1, ldsA, ldsB);
        __syncthreads();
        #pragma unroll 1
        for (int e = tid; e < KD * KD; e += TPB) {
            const int i = e >> 8, j = e & 255;
            P[e] = PRED[e] - 0.5f * (W1[e] + W1[j * KD + i]);
        }
        __syncthreads();
    }

    if (tid == 0) out[0] = nll_acc;
}

extern "C" void kernel_launch(void* const* d_in, const int* in_sizes, int n_in,
                              void* d_out, int out_size, void* d_ws, size_t ws_size,
                              hipStream_t stream)
{
    (void)n_in; (void)out_size; (void)ws_size;
    const float* obs   = (const float*)d_in[0];
    const float* trans = (const float*)d_in[1];
    const float* forc  = (const float*)d_in[2];
    const float* pcov  = (const float*)d_in[3];
    const float* mcov  = (const float*)d_in[4];
    const float* imean = (const float*)d_in[5];
    const float* icovd = (const float*)d_in[6];
    const int T = in_sizes[0] / KD;   // observations is T*KD

    kf_scan_kernel<<<1, TPB, 0, stream>>>(obs, trans, forc, pcov, mcov,
                                          imean, icovd,
                                          (float*)d_out, (float*)d_ws, T);
}